// MyGaussianModel_44238163148812
// MI455X (gfx1250) — compile-verified
//
#include <hip/hip_runtime.h>
#include <cstdint>
#include <cstddef>

// ---------------------------------------------------------------------------
// Mesh-bound Gaussian flattening for MI455X (gfx1250, wave32).
// Bandwidth-bound (~1.8 FLOP/B): strategy is pure data movement.
//   Phase 1: per-face frame/center/scale -> 64B records in d_ws
//            (649 KB, permanently L2-resident out of 192 MB).
//   Phase 2: per-gaussian gather (RT-cached) + streamed inputs (TH_LOAD_NT)
//            + 3x3 algebra; outputs staged in LDS and drained with
//            global_store_async_from_lds_b128 th:TH_STORE_NT (ASYNCcnt path)
//            so the 104 MB write-once stream does not evict the face table.
// ---------------------------------------------------------------------------

typedef float v4f __attribute__((ext_vector_type(4)));

__device__ __forceinline__ void face_basis(
    const float* __restrict__ verts, const int* __restrict__ faces, int f,
    float4& r0, float4& r1, float4& r2, float& fs)
{
    const int i0 = faces[3 * f + 0];
    const int i1 = faces[3 * f + 1];
    const int i2 = faces[3 * f + 2];
    const float v0x = verts[3 * i0 + 0], v0y = verts[3 * i0 + 1], v0z = verts[3 * i0 + 2];
    const float v1x = verts[3 * i1 + 0], v1y = verts[3 * i1 + 1], v1z = verts[3 * i1 + 2];
    const float v2x = verts[3 * i2 + 0], v2y = verts[3 * i2 + 1], v2z = verts[3 * i2 + 2];

    const float e01x = v1x - v0x, e01y = v1y - v0y, e01z = v1z - v0z;
    const float e02x = v2x - v0x, e02y = v2y - v0y, e02z = v2z - v0z;

    const float d01 = e01x * e01x + e01y * e01y + e01z * e01z;
    const float in0 = rsqrtf(fmaxf(d01, 1e-20f));
    const float a0x = e01x * in0, a0y = e01y * in0, a0z = e01z * in0;

    // a1 = normalize(cross(a0, e02))
    const float c1x = a0y * e02z - a0z * e02y;
    const float c1y = a0z * e02x - a0x * e02z;
    const float c1z = a0x * e02y - a0y * e02x;
    const float in1 = rsqrtf(fmaxf(c1x * c1x + c1y * c1y + c1z * c1z, 1e-20f));
    const float a1x = c1x * in1, a1y = c1y * in1, a1z = c1z * in1;

    // a2 = -normalize(cross(a1, a0))
    const float c2x = a1y * a0z - a1z * a0y;
    const float c2y = a1z * a0x - a1x * a0z;
    const float c2z = a1x * a0y - a1y * a0x;
    const float in2 = rsqrtf(fmaxf(c2x * c2x + c2y * c2y + c2z * c2z, 1e-20f));
    const float a2x = -c2x * in2, a2y = -c2y * in2, a2z = -c2z * in2;

    const float s0 = sqrtf(d01);
    const float s1 = fabsf(a2x * e02x + a2y * e02y + a2z * e02z);
    fs = 0.5f * (s0 + s1);

    const float third = 1.0f / 3.0f;
    // Rows of O = [a0 | a1 | a2], with face_center packed in .w
    r0 = make_float4(a0x, a1x, a2x, (v0x + v1x + v2x) * third);
    r1 = make_float4(a0y, a1y, a2y, (v0y + v1y + v2y) * third);
    r2 = make_float4(a0z, a1z, a2z, (v0z + v1z + v2z) * third);
}

__global__ __launch_bounds__(256) void face_kernel(
    const float* __restrict__ verts, const int* __restrict__ faces,
    float* __restrict__ rec, int F)
{
    const int f = (int)(blockIdx.x * 256u) + (int)threadIdx.x;
    if (f >= F) return;
    float4 r0, r1, r2; float fs;
    face_basis(verts, faces, f, r0, r1, r2, fs);
    float4* r = reinterpret_cast<float4*>(rec + (size_t)f * 16);
    r[0] = r0;
    r[1] = r1;
    r[2] = r2;
    r[3] = make_float4(fs, 0.f, 0.f, 0.f);
}

template <bool FUSED>
__global__ __launch_bounds__(256) void splat_kernel(
    const int* __restrict__ binding, const float* __restrict__ xyz,
    const float* __restrict__ rot, const float* __restrict__ scal,
    const float* __restrict__ opac, const float* __restrict__ fdc,
    const float* __restrict__ rec,
    const float* __restrict__ verts, const int* __restrict__ faces,
    float* __restrict__ outp, int N)
{
    __shared__ float smem[256 * 13];   // 13,312 B staging tile
    const int tid = (int)threadIdx.x;
    const int gid = (int)(blockIdx.x * 256u) + tid;
    const bool full_block = ((blockIdx.x + 1u) * 256u <= (unsigned)N);

    float o[13];
    if (!full_block) {
        #pragma unroll
        for (int k = 0; k < 13; ++k) o[k] = 0.0f;
    }

    if (gid < N) {
        const size_t g = (size_t)gid;
        const int b = __builtin_nontemporal_load(binding + g);
        const float* rb = rec + (size_t)b * 16;
        if (!FUSED) __builtin_prefetch(rb, 0, 3);  // global_prefetch_b8 on gathered record

        // Streaming per-gaussian inputs: read-once -> non-temporal loads
        // (th:TH_LOAD_NT) so L2 stays reserved for the face-record table.
        const v4f q4 = __builtin_nontemporal_load((const v4f*)(rot + g * 4));
        const float px  = __builtin_nontemporal_load(xyz  + g * 3 + 0);
        const float py  = __builtin_nontemporal_load(xyz  + g * 3 + 1);
        const float pz  = __builtin_nontemporal_load(xyz  + g * 3 + 2);
        const float sc0 = __builtin_nontemporal_load(scal + g * 3 + 0);
        const float sc1 = __builtin_nontemporal_load(scal + g * 3 + 1);
        const float sc2 = __builtin_nontemporal_load(scal + g * 3 + 2);
        const float opv = __builtin_nontemporal_load(opac + g);
        const float d0  = __builtin_nontemporal_load(fdc  + g * 3 + 0);
        const float d1  = __builtin_nontemporal_load(fdc  + g * 3 + 1);
        const float d2  = __builtin_nontemporal_load(fdc  + g * 3 + 2);

        float4 r0, r1, r2; float fs;
        if (FUSED) {
            face_basis(verts, faces, b, r0, r1, r2, fs);
        } else {
            r0 = *reinterpret_cast<const float4*>(rb + 0);   // RT: keep cached
            r1 = *reinterpret_cast<const float4*>(rb + 4);
            r2 = *reinterpret_cast<const float4*>(rb + 8);
            fs = rb[12];
        }

        // xyz_out = (O @ xyz) * fs + center
        o[0] = (r0.x * px + r0.y * py + r0.z * pz) * fs + r0.w;
        o[1] = (r1.x * px + r1.y * py + r1.z * pz) * fs + r1.w;
        o[2] = (r2.x * px + r2.y * py + r2.z * pz) * fs + r2.w;

        // color = clip(concat(0.5 + 0.282*dc, sigmoid(op)) * 255, 0, 255)
        o[3] = fminf(fmaxf((0.5f + 0.282f * d0) * 255.0f, 0.0f), 255.0f);
        o[4] = fminf(fmaxf((0.5f + 0.282f * d1) * 255.0f, 0.0f), 255.0f);
        o[5] = fminf(fmaxf((0.5f + 0.282f * d2) * 255.0f, 0.0f), 255.0f);
        const float sg = 1.0f / (1.0f + expf(-opv));
        o[6] = fminf(fmaxf(sg * 255.0f, 0.0f), 255.0f);

        // Quaternion (wxyz) -> rotation matrix R
        float qw = q4.x, qx = q4.y, qy = q4.z, qz = q4.w;
        const float qi = 1.0f / (sqrtf(qw * qw + qx * qx + qy * qy + qz * qz) + 1e-12f);
        qw *= qi; qx *= qi; qy *= qi; qz *= qi;
        const float R00 = 1.0f - 2.0f * (qy * qy + qz * qz);
        const float R01 = 2.0f * (qx * qy - qz * qw);
        const float R02 = 2.0f * (qx * qz + qy * qw);
        const float R10 = 2.0f * (qx * qy + qz * qw);
        const float R11 = 1.0f - 2.0f * (qx * qx + qz * qz);
        const float R12 = 2.0f * (qy * qz - qx * qw);
        const float R20 = 2.0f * (qx * qz - qy * qw);
        const float R21 = 2.0f * (qy * qz + qx * qw);
        const float R22 = 1.0f - 2.0f * (qx * qx + qy * qy);

        // M = O @ R   (rows of O are r0/r1/r2 .xyz)
        const float m00 = r0.x * R00 + r0.y * R10 + r0.z * R20;
        const float m01 = r0.x * R01 + r0.y * R11 + r0.z * R21;
        const float m02 = r0.x * R02 + r0.y * R12 + r0.z * R22;
        const float m10 = r1.x * R00 + r1.y * R10 + r1.z * R20;
        const float m11 = r1.x * R01 + r1.y * R11 + r1.z * R21;
        const float m12 = r1.x * R02 + r1.y * R12 + r1.z * R22;
        const float m20 = r2.x * R00 + r2.y * R10 + r2.z * R20;
        const float m21 = r2.x * R01 + r2.y * R11 + r2.z * R21;
        const float m22 = r2.x * R02 + r2.y * R12 + r2.z * R22;

        // scales = exp(scaling) * fs ;  C[j][k] = sum_i M[j][i]*M[k][i]*scales[i]^2
        const float e0 = expf(sc0) * fs;
        const float e1 = expf(sc1) * fs;
        const float e2 = expf(sc2) * fs;
        const float w00 = m00 * e0, w01 = m01 * e1, w02 = m02 * e2;
        const float w10 = m10 * e0, w11 = m11 * e1, w12 = m12 * e2;
        const float w20 = m20 * e0, w21 = m21 * e1, w22 = m22 * e2;
        o[7]  = w00 * w00 + w01 * w01 + w02 * w02;   // C00
        o[8]  = w00 * w10 + w01 * w11 + w02 * w12;   // C01
        o[9]  = w00 * w20 + w01 * w21 + w02 * w22;   // C02
        o[10] = w10 * w10 + w11 * w11 + w12 * w12;   // C11
        o[11] = w10 * w20 + w11 * w21 + w12 * w22;   // C12
        o[12] = w20 * w20 + w21 * w21 + w22 * w22;   // C22
    }

    if (full_block) {
        // Stage 13-float rows in LDS (stride 13 is coprime with 64 banks ->
        // conflict-free), then drain the 13,312B tile with coalesced async
        // b128 LDS->global DMA stores (ASYNCcnt path). TH_STORE_NT: the
        // output is write-once, keep it from flooding L2.
        #pragma unroll
        for (int k = 0; k < 13; ++k) smem[tid * 13 + k] = o[k];
        __syncthreads();   // LDS writes visible before async engine reads

        const uint32_t lds0 = (uint32_t)(size_t)(&smem[0]);  // LDS byte offset
        const uint32_t base = blockIdx.x * (256u * 13u * 4u);
        for (int i = tid; i < 832; i += 256) {               // 832 * 16B = tile
            const uint32_t goff = base + (uint32_t)i * 16u;
            const uint32_t loff = lds0 + (uint32_t)i * 16u;
            asm volatile("global_store_async_from_lds_b128 %0, %1, %2 th:TH_STORE_NT"
                         :: "v"(goff), "v"(loff), "s"(outp)
                         : "memory");
        }
        asm volatile("s_wait_asynccnt 0" ::: "memory");
    } else if (gid < N) {
        float* p = outp + (size_t)gid * 13;
        #pragma unroll
        for (int k = 0; k < 13; ++k) p[k] = o[k];
    }
}

extern "C" void kernel_launch(void* const* d_in, const int* in_sizes, int n_in,
                              void* d_out, int out_size, void* d_ws, size_t ws_size,
                              hipStream_t stream)
{
    const float* verts   = (const float*)d_in[0];
    const int*   faces   = (const int*)d_in[1];
    const int*   binding = (const int*)d_in[2];
    const float* xyz     = (const float*)d_in[3];
    const float* rot     = (const float*)d_in[4];
    const float* scal    = (const float*)d_in[5];
    const float* opac    = (const float*)d_in[6];
    const float* fdc     = (const float*)d_in[7];
    // d_in[8] (features_rest) is unused by the reference: never touched.

    const int F = in_sizes[1] / 3;
    const int N = in_sizes[2];
    float* rec = (float*)d_ws;
    const size_t rec_bytes = (size_t)F * 16 * sizeof(float);

    const int gblocks = (N + 255) / 256;
    if (d_ws != nullptr && ws_size >= rec_bytes) {
        face_kernel<<<(F + 255) / 256, 256, 0, stream>>>(verts, faces, rec, F);
        splat_kernel<false><<<gblocks, 256, 0, stream>>>(
            binding, xyz, rot, scal, opac, fdc, rec, verts, faces,
            (float*)d_out, N);
    } else {
        // Workspace too small: fused fallback recomputes face math per point.
        splat_kernel<true><<<gblocks, 256, 0, stream>>>(
            binding, xyz, rot, scal, opac, fdc, nullptr, verts, faces,
            (float*)d_out, N);
    }
}